// GATFormal_30030411334391
// MI455X (gfx1250) — compile-verified
//
#include <hip/hip_runtime.h>
#include <hip/hip_bf16.h>
#include <math.h>

typedef __attribute__((ext_vector_type(2))) float v2f;
typedef __attribute__((ext_vector_type(8))) float v8f;

#define N_NODES 50000
#define E_RAW   400000
#define E_TOT   450000

// ---------------------------------------------------------------------------
// FP32 WMMA GEMM: C[M,N] = A[M,K] * B[K,N].  One 16x16 tile per wave,
// K stepped by 4 via V_WMMA_F32_16X16X4_F32. M,N multiples of 16; K of 4.
// ---------------------------------------------------------------------------
__global__ void k_gemm_wmma_f32(const float* __restrict__ A,
                                const float* __restrict__ B,
                                float* __restrict__ C,
                                int M, int N, int K) {
    const int lane   = threadIdx.x & 31;
    const int wave   = threadIdx.x >> 5;
    const int tilesN = N >> 4;
    const int tilesM = M >> 4;
    const int tile   = blockIdx.x * (blockDim.x >> 5) + wave;
    if (tile >= tilesM * tilesN) return;          // wave-uniform exit

    const int tm = (tile / tilesN) << 4;
    const int tn = (tile % tilesN) << 4;

    const int r  = lane & 15;          // row (A) / col (B,C) within tile
    const int kk = (lane >> 4) << 1;   // lane-group K offset: 0 or 2

    const float* __restrict__ Arow = A + (size_t)(tm + r) * K;
    const float* __restrict__ Bcol = B + (tn + r);

    v8f c = {};
    for (int k = 0; k < K; k += 4) {
        v2f a, b;
        a.x = Arow[k + kk];
        a.y = Arow[k + kk + 1];
        b.x = Bcol[(size_t)(k + kk) * N];
        b.y = Bcol[(size_t)(k + kk + 1) * N];
        c = __builtin_amdgcn_wmma_f32_16x16x4_f32(
                /*neg_a=*/false, a, /*neg_b=*/false, b,
                /*c_mod=*/(short)0, c, /*reuse_a=*/false, /*reuse_b=*/false);
    }

    // C/D layout: VGPR v -> M = v (+8 for lanes 16-31), N = lane&15
    const int row0 = tm + ((lane >> 4) << 3);
    const int col  = tn + (lane & 15);
#pragma unroll
    for (int v = 0; v < 8; ++v)
        C[(size_t)(row0 + v) * N + col] = c[v];
}

// ---------------------------------------------------------------------------
// Per-node attention scalars: out_s[i] = h[i,:].a_s ; out_d[i] = h[i,:].a_d
// One wave per node, wave32 shuffle reduction.
// ---------------------------------------------------------------------------
__global__ void k_node_dots(const float* __restrict__ h,
                            const float* __restrict__ a_s,
                            const float* __restrict__ a_d,
                            float* __restrict__ out_s,
                            float* __restrict__ out_d,
                            int N, int F) {
    const int node = blockIdx.x * (blockDim.x >> 5) + (threadIdx.x >> 5);
    const int lane = threadIdx.x & 31;
    if (node >= N) return;
    const float* __restrict__ row = h + (size_t)node * F;
    float ss = 0.f, sd = 0.f;
    for (int j = lane; j < F; j += 32) {
        float v = row[j];
        ss += v * a_s[j];
        sd += v * a_d[j];
    }
#pragma unroll
    for (int off = 16; off > 0; off >>= 1) {
        ss += __shfl_down(ss, off, 32);
        sd += __shfl_down(sd, off, 32);
    }
    if (lane == 0) { out_s[node] = ss; out_d[node] = sd; }
}

// ---------------------------------------------------------------------------
// Fill helper (also used to zero the uint segment-max buffer: 0u < enc(-inf))
// ---------------------------------------------------------------------------
__global__ void k_fill(float* __restrict__ p, float v, size_t n) {
    size_t i = (size_t)blockIdx.x * blockDim.x + threadIdx.x;
    if (i < n) p[i] = v;
}

__device__ __forceinline__ void edge_sd(const int* __restrict__ eidx, int e,
                                        int& s, int& d) {
    if (e < E_RAW) { s = eidx[e]; d = eidx[E_RAW + e]; }
    else           { s = d = e - E_RAW; }     // appended self-loops
}

__device__ __forceinline__ unsigned enc_f32(float x) {
    unsigned u = __float_as_uint(x);
    return (u & 0x80000000u) ? ~u : (u | 0x80000000u);
}
__device__ __forceinline__ float dec_f32(unsigned u) {
    return __uint_as_float((u & 0x80000000u) ? (u & 0x7fffffffu) : ~u);
}

// leaky_relu logits + segment-max (order-preserving uint atomicMax)
__global__ void k_edge_logits(const int* __restrict__ eidx,
                              const float* __restrict__ as,
                              const float* __restrict__ ad,
                              float* __restrict__ logits,
                              unsigned* __restrict__ menc) {
    int e = blockIdx.x * blockDim.x + threadIdx.x;
    if (e >= E_TOT) return;
    int s, d; edge_sd(eidx, e, s, d);
    float l = as[s] + ad[d];
    l = (l > 0.f) ? l : 0.2f * l;
    logits[e] = l;
    atomicMax(&menc[d], enc_f32(l));
}

// w = exp(logit - max[dst]); segment-sum into denom.  In-place on logits buf.
__global__ void k_edge_expw(const int* __restrict__ eidx,
                            float* __restrict__ ew,
                            const unsigned* __restrict__ menc,
                            float* __restrict__ denom) {
    int e = blockIdx.x * blockDim.x + threadIdx.x;
    if (e >= E_TOT) return;
    int s, d; edge_sd(eidx, e, s, d); (void)s;
    float we = __expf(ew[e] - dec_f32(menc[d]));
    ew[e] = we;
    atomicAdd(&denom[d], we);
}

// agg[dst,:] += h[src,:] * (w/denom[dst]).  One wave per edge, fp32 atomics.
__global__ void k_edge_aggregate(const int* __restrict__ eidx,
                                 const float* __restrict__ h,
                                 const float* __restrict__ ew,
                                 const float* __restrict__ denom,
                                 float* __restrict__ agg, int F) {
    const int e    = blockIdx.x * (blockDim.x >> 5) + (threadIdx.x >> 5);
    const int lane = threadIdx.x & 31;
    if (e >= E_TOT) return;
    int s, d; edge_sd(eidx, e, s, d);
    const float coeff = ew[e] / denom[d];
    const float* __restrict__ hr = h   + (size_t)s * F;
    float* __restrict__       ar = agg + (size_t)d * F;
    for (int j = lane; j < F; j += 32)
        atomicAdd(&ar[j], hr[j] * coeff);
}

__global__ void k_bias_relu(float* __restrict__ x, const float* __restrict__ b,
                            int N, int F, int do_relu) {
    size_t i = (size_t)blockIdx.x * blockDim.x + threadIdx.x;
    if (i >= (size_t)N * F) return;
    float v = x[i] + b[i % F];
    x[i] = do_relu ? fmaxf(v, 0.f) : v;
}

// final: softmax(agg + b3) over 16 classes, in place in d_out
__global__ void k_softmax16(float* __restrict__ io, const float* __restrict__ b,
                            int N) {
    int i = blockIdx.x * blockDim.x + threadIdx.x;
    if (i >= N) return;
    float v[16], m = -INFINITY;
#pragma unroll
    for (int j = 0; j < 16; ++j) { v[j] = io[i * 16 + j] + b[j]; m = fmaxf(m, v[j]); }
    float sum = 0.f;
#pragma unroll
    for (int j = 0; j < 16; ++j) { v[j] = __expf(v[j] - m); sum += v[j]; }
    float inv = 1.f / sum;
#pragma unroll
    for (int j = 0; j < 16; ++j) io[i * 16 + j] = v[j] * inv;
}

// ---------------------------------------------------------------------------
static inline int cdiv(long long a, long long b) { return (int)((a + b - 1) / b); }

extern "C" void kernel_launch(void* const* d_in, const int* in_sizes, int n_in,
                              void* d_out, int out_size, void* d_ws, size_t ws_size,
                              hipStream_t stream) {
    const float* x    = (const float*)d_in[0];
    const int*   eidx = (const int*)  d_in[1];   // [2, 400000] int32
    const float* W1  = (const float*)d_in[2];
    const float* as1 = (const float*)d_in[3];
    const float* ad1 = (const float*)d_in[4];
    const float* b1  = (const float*)d_in[5];
    const float* W2  = (const float*)d_in[6];
    const float* as2 = (const float*)d_in[7];
    const float* ad2 = (const float*)d_in[8];
    const float* b2  = (const float*)d_in[9];
    const float* W3  = (const float*)d_in[10];
    const float* as3 = (const float*)d_in[11];
    const float* ad3 = (const float*)d_in[12];
    const float* b3  = (const float*)d_in[13];

    const int N = N_NODES;
    const int F1 = 512, F2 = 256, F3 = 16;

    // workspace carve-up (floats)
    float* ws      = (float*)d_ws;
    float* buf_h   = ws;                                 // N*512 (gemm out)
    float* buf_a   = buf_h + (size_t)N * F1;             // N*512 (aggregate)
    float* alpha_s = buf_a + (size_t)N * F1;             // N
    float* alpha_d = alpha_s + N;                        // N
    unsigned* menc = (unsigned*)(alpha_d + N);           // N
    float* denom   = (float*)(menc + N);                 // N
    float* ewbuf   = denom + N;                          // E_TOT (logits -> w)

    const int TB = 256;                                  // 8 waves/block
    const int edge_blocks  = cdiv(E_TOT, TB);
    const int eagg_blocks  = cdiv(E_TOT, TB / 32);
    const int node_blocks  = cdiv(N, TB / 32);

    auto run_layer = [&](const float* hin, const float* W, const float* avs,
                         const float* avd, const float* bias, float* hbuf,
                         float* aggbuf, int Fin, int Fout, int relu) {
        // h = hin @ W   (WMMA fp32)
        {
            int tiles = (N / 16) * (Fout / 16);
            k_gemm_wmma_f32<<<cdiv(tiles, TB / 32), TB, 0, stream>>>(
                hin, W, hbuf, N, Fout, Fin);
        }
        // per-node attention scalars
        k_node_dots<<<node_blocks, TB, 0, stream>>>(hbuf, avs, avd,
                                                    alpha_s, alpha_d, N, Fout);
        // zero segment buffers + aggregate target
        k_fill<<<cdiv(N, TB), TB, 0, stream>>>((float*)menc, 0.0f, (size_t)N);
        k_fill<<<cdiv(N, TB), TB, 0, stream>>>(denom, 0.0f, (size_t)N);
        k_fill<<<cdiv((size_t)N * Fout, TB), TB, 0, stream>>>(aggbuf, 0.0f,
                                                              (size_t)N * Fout);
        // segment softmax + weighted scatter-aggregate
        k_edge_logits<<<edge_blocks, TB, 0, stream>>>(eidx, alpha_s, alpha_d,
                                                      ewbuf, menc);
        k_edge_expw<<<edge_blocks, TB, 0, stream>>>(eidx, ewbuf, menc, denom);
        k_edge_aggregate<<<eagg_blocks, TB, 0, stream>>>(eidx, hbuf, ewbuf,
                                                         denom, aggbuf, Fout);
        if (relu)
            k_bias_relu<<<cdiv((size_t)N * Fout, TB), TB, 0, stream>>>(
                aggbuf, bias, N, Fout, 1);
    };

    // Layer 1: x[ N,128 ] -> buf_a[ N,512 ] (relu)
    run_layer(x, W1, as1, ad1, b1, buf_h, buf_a, 128, F1, 1);
    // Layer 2: buf_a[ N,512 ] -> buf_a[ N,256 ] (relu)   (hbuf reused)
    run_layer(buf_a, W2, as2, ad2, b2, buf_h, buf_a, F1, F2, 1);
    // Layer 3: buf_a[ N,256 ] -> d_out[ N,16 ], bias+softmax fused at end
    run_layer(buf_a, W3, as3, ad3, b3, buf_h, (float*)d_out, F2, F3, 0);
    k_softmax16<<<cdiv(N, TB), TB, 0, stream>>>((float*)d_out, b3, N);
}